// KGATConv_30846455120404
// MI455X (gfx1250) — compile-verified
//
#include <hip/hip_runtime.h>
#include <math.h>

#define D 64
#define OUTSTRIDE 192  // 3*D

typedef __attribute__((ext_vector_type(2))) float v2f;
typedef __attribute__((ext_vector_type(4))) float v4f;
typedef __attribute__((ext_vector_type(8))) float v8f;

// ---------------------------------------------------------------------------
// out[:, 0:64] = embeddings
// ---------------------------------------------------------------------------
__global__ __launch_bounds__(256) void copy_embed_kernel(
    const float* __restrict__ emb, float* __restrict__ out, int n) {
  int gid = blockIdx.x * 256 + threadIdx.x;
  int m = gid >> 6;
  int c = gid & 63;
  if (m < n) out[(size_t)m * OUTSTRIDE + c] = emb[(size_t)m * D + c];
}

// ---------------------------------------------------------------------------
// zero a float buffer
// ---------------------------------------------------------------------------
__global__ __launch_bounds__(256) void zero_kernel(float* __restrict__ p, size_t count) {
  size_t gid = (size_t)blockIdx.x * 256 + threadIdx.x;
  if (gid < count) p[gid] = 0.0f;
}

// ---------------------------------------------------------------------------
// SpMM: acc[src[e], :] += x[dst[e], :] * val[e]
// 16 lanes per edge, one float4 (16B) per lane -> a 256B row is one coalesced
// cacheline-pair gather. 4 relaxed agent-scope f32 atomic adds per lane
// (non-returning global_atomic_add_f32).
// ---------------------------------------------------------------------------
__global__ __launch_bounds__(256) void spmm_kernel(
    const float* __restrict__ x, const float* __restrict__ edge_val,
    const int* __restrict__ edge_src, const int* __restrict__ edge_dst,
    float* __restrict__ acc, int n_edges) {
  long gid = (long)blockIdx.x * 256 + threadIdx.x;
  int e = (int)(gid >> 4);
  int c = ((int)gid & 15) * 4;
  if (e >= n_edges) return;
  int src = edge_src[e];
  int dst = edge_dst[e];
  float v = edge_val[e];
  const float4 xv = *reinterpret_cast<const float4*>(x + (size_t)dst * D + c);
  float* p = acc + (size_t)src * D + c;
  __hip_atomic_fetch_add(p + 0, xv.x * v, __ATOMIC_RELAXED, __HIP_MEMORY_SCOPE_AGENT);
  __hip_atomic_fetch_add(p + 1, xv.y * v, __ATOMIC_RELAXED, __HIP_MEMORY_SCOPE_AGENT);
  __hip_atomic_fetch_add(p + 2, xv.z * v, __ATOMIC_RELAXED, __HIP_MEMORY_SCOPE_AGENT);
  __hip_atomic_fetch_add(p + 3, xv.w * v, __ATOMIC_RELAXED, __HIP_MEMORY_SCOPE_AGENT);
}

// ---------------------------------------------------------------------------
// BiCombiner + L2-normalize, fp32 WMMA (V_WMMA_F32_16X16X4_F32).
// One wave computes a 16x64 output panel:
//   ego_next = leaky((ego+side)@Ws + bs) + leaky((ego*side)@Wb + bb)
//   out slice = l2norm(ego_next)
// 8 waves / block -> 128 rows per block.
//
// Weights are staged in LDS pre-swizzled for the WMMA 4x16 B-fragment layout:
//   lW[kp*64 + c] = {Ws[2kp][c], Ws[2kp+1][c], Wb[2kp][c], Wb[2kp+1][c]}
// so each (k-step, col-tile) fragment pair is ONE ds_load_b128 landing in
// adjacent even-aligned VGPRs (no shuffle movs).
// ---------------------------------------------------------------------------
__global__ __launch_bounds__(256) void combiner_kernel(
    const float* __restrict__ ego_in, const float* __restrict__ side,
    const float* __restrict__ Wsum, const float* __restrict__ bsum,
    const float* __restrict__ Wbi, const float* __restrict__ bbi,
    float* __restrict__ ego_out,  // may be nullptr (last layer)
    float* __restrict__ out, int col_off, int n) {
  // 32 k-pairs x 64 cols x float4 = 32 KB
  __shared__ v4f lW[(D / 2) * D];
  int tid = threadIdx.x;
  // 2048 float4 elements, 256 threads -> 8 each; coalesced in c.
  for (int i = tid; i < (D / 2) * D; i += 256) {
    int kp = i >> 6;       // k-pair index: k = 2*kp, 2*kp+1
    int c = i & 63;        // output column
    v4f w;
    w.x = Wsum[(2 * kp) * D + c];
    w.y = Wsum[(2 * kp + 1) * D + c];
    w.z = Wbi[(2 * kp) * D + c];
    w.w = Wbi[(2 * kp + 1) * D + c];
    lW[i] = w;
  }
  __syncthreads();

  int wave = tid >> 5;
  int lane = tid & 31;
  int half = lane >> 4;  // 0: K%4 in {0,1} / rows 0-7 of C ; 1: K%4 in {2,3} / rows 8-15
  int l = lane & 15;
  long rowbase = ((long)blockIdx.x * 8 + wave) * 16;

  // Clamp load row so EXEC stays all-1s (WMMA requirement); stores are guarded.
  long mload = rowbase + l;
  if (mload > (long)n - 1) mload = (long)n - 1;
  if (mload < 0) mload = 0;
  const float* egoRow = ego_in + mload * D;
  const float* sideRow = side + mload * D;

  v8f csum[4] = {};  // (ego+side) @ Wsum, 4 column tiles of 16
  v8f cbi[4] = {};   // (ego*side) @ Wbi

#pragma unroll
  for (int kk = 0; kk < 16; ++kk) {
    // A fragment layout (32-bit A, 16x4): lanes 0-15 hold K=4kk..4kk+1,
    // lanes 16-31 hold K=4kk+2..4kk+3, M = lane&15.
    int k0 = kk * 4 + half * 2;
    int kp = k0 >> 1;  // k-pair index into lW
    float e0 = egoRow[k0], e1 = egoRow[k0 + 1];
    float s0 = sideRow[k0], s1 = sideRow[k0 + 1];
    v2f aS = {e0 + s0, e1 + s1};
    v2f aB = {e0 * s0, e1 * s1};
#pragma unroll
    for (int nt = 0; nt < 4; ++nt) {
      int cb = nt * 16 + l;
      v4f w = lW[kp * D + cb];  // one ds_load_b128: both fragments
      v2f bS = {w.x, w.y};
      v2f bB = {w.z, w.w};
      csum[nt] = __builtin_amdgcn_wmma_f32_16x16x4_f32(
          false, aS, false, bS, (short)0, csum[nt], false, false);
      cbi[nt] = __builtin_amdgcn_wmma_f32_16x16x4_f32(
          false, aB, false, bB, (short)0, cbi[nt], false, false);
    }
  }

  // Epilogue: bias + leaky + add, accumulate squared sums per row.
  // C/D layout: VGPR r holds row (rowbase + 8*half + r), col = nt*16 + (lane&15).
  float val[4][8];
  float ss[8];
#pragma unroll
  for (int r = 0; r < 8; ++r) ss[r] = 0.0f;
#pragma unroll
  for (int nt = 0; nt < 4; ++nt) {
    int cb = nt * 16 + l;
    float bsv = bsum[cb];
    float bbv = bbi[cb];
#pragma unroll
    for (int r = 0; r < 8; ++r) {
      float x1 = csum[nt][r] + bsv;
      x1 = x1 > 0.0f ? x1 : 0.01f * x1;
      float x2 = cbi[nt][r] + bbv;
      x2 = x2 > 0.0f ? x2 : 0.01f * x2;
      float v = x1 + x2;
      val[nt][r] = v;
      ss[r] += v * v;
    }
  }
  // Row sums live entirely within one 16-lane half: butterfly with masks < 16.
#pragma unroll
  for (int r = 0; r < 8; ++r) {
    float s = ss[r];
    s += __shfl_xor(s, 1);
    s += __shfl_xor(s, 2);
    s += __shfl_xor(s, 4);
    s += __shfl_xor(s, 8);
    float nrm = sqrtf(s);
    ss[r] = 1.0f / fmaxf(nrm, 1e-12f);
  }

#pragma unroll
  for (int r = 0; r < 8; ++r) {
    long m = rowbase + half * 8 + r;
    if (m < n) {
      float inv = ss[r];
#pragma unroll
      for (int nt = 0; nt < 4; ++nt) {
        int cb = nt * 16 + l;
        float v = val[nt][r];
        if (ego_out) ego_out[m * D + cb] = v;
        out[m * OUTSTRIDE + col_off + cb] = v * inv;
      }
    }
  }
}

// ---------------------------------------------------------------------------
extern "C" void kernel_launch(void* const* d_in, const int* in_sizes, int n_in,
                              void* d_out, int out_size, void* d_ws, size_t ws_size,
                              hipStream_t stream) {
  const float* emb = (const float*)d_in[0];
  const float* eval_ = (const float*)d_in[1];
  const float* Ws0 = (const float*)d_in[2];
  const float* bs0 = (const float*)d_in[3];
  const float* Wb0 = (const float*)d_in[4];
  const float* bb0 = (const float*)d_in[5];
  const float* Ws1 = (const float*)d_in[6];
  const float* bs1 = (const float*)d_in[7];
  const float* Wb1 = (const float*)d_in[8];
  const float* bb1 = (const float*)d_in[9];
  const int* esrc = (const int*)d_in[10];
  const int* edst = (const int*)d_in[11];

  int n = in_sizes[0] / D;
  int E = in_sizes[1];
  float* out = (float*)d_out;

  float* acc = (float*)d_ws;                 // [n, 64] spmm accumulator
  float* ego = acc + (size_t)n * D;          // [n, 64] layer-1 node features

  dim3 blk(256);
  size_t accN = (size_t)n * D;
  int zgrid = (int)((accN + 255) / 256);
  int sgrid = (int)(((size_t)E * 16 + 255) / 256);
  int cgrid = (n + 127) / 128;
  int cpgrid = (int)(((size_t)n * D + 255) / 256);

  // out[:, 0:64] = embeddings
  copy_embed_kernel<<<cpgrid, blk, 0, stream>>>(emb, out, n);

  // ---- layer 1 ----
  zero_kernel<<<zgrid, blk, 0, stream>>>(acc, accN);
  spmm_kernel<<<sgrid, blk, 0, stream>>>(emb, eval_, esrc, edst, acc, E);
  combiner_kernel<<<cgrid, blk, 0, stream>>>(emb, acc, Ws0, bs0, Wb0, bb0,
                                             ego, out, /*col_off=*/64, n);

  // ---- layer 2 ----
  zero_kernel<<<zgrid, blk, 0, stream>>>(acc, accN);
  spmm_kernel<<<sgrid, blk, 0, stream>>>(ego, eval_, esrc, edst, acc, E);
  combiner_kernel<<<cgrid, blk, 0, stream>>>(ego, acc, Ws1, bs1, Wb1, bb1,
                                             nullptr, out, /*col_off=*/128, n);
}